// GCNModelGumbel_2_13804024889380
// MI455X (gfx1250) — compile-verified
//
#include <hip/hip_runtime.h>
#include <math.h>

// ---- CDNA5 WMMA vector types -------------------------------------------------
typedef __attribute__((ext_vector_type(2))) float v2f;   // A/B frag of 16x16x4 f32
typedef __attribute__((ext_vector_type(8))) float v8f;   // 16x16 f32 C/D frag

#define DIM       128   // embedding dim
#define KCAT      64    // categorical dim
#define NNEG      5
#define WAVES_WG  8
#define ROWS_WG   (WAVES_WG * 16)   // 128 batch rows per workgroup

__device__ __forceinline__ float log_sigmoid(float x) {
    // stable: min(x,0) - log1p(exp(-|x|))
    float ax = fabsf(x);
    return fminf(x, 0.0f) - log1pf(__expf(-ax));
}

__global__ __launch_bounds__(256)
void gcn_gumbel_main(const int*   __restrict__ w_idx,
                     const int*   __restrict__ c_idx,
                     const int*   __restrict__ neg_idx,
                     const int*   __restrict__ temp_p,
                     const float* __restrict__ gumbel,
                     const float* __restrict__ node_emb,
                     const float* __restrict__ ctx_emb,
                     const float* __restrict__ comm_w,
                     float*       __restrict__ loss_partials,
                     float*       __restrict__ out_q,
                     float*       __restrict__ out_prior)
{
    // LDS layout (floats):
    //   [0      , 8192 )  community_w, row-major [64][128]          (32 KB)
    //   [8192   , 24576)  per-wave w_e tiles, 8 x (16x128)          (64 KB)
    //   [24576  , 40960)  per-wave (w_e*c_e) tiles, 8 x (16x128)    (64 KB)
    // After the GEMM the per-wave tiles are dead and are reused as
    // [16][64] row-major result tiles (prior logits / q logits).
    __shared__ float smem[40960];     // 160 KB
    __shared__ float red[256];

    const int tid  = threadIdx.x;
    const int wave = tid >> 5;
    const int lane = tid & 31;

    float* sW  = smem;                                  // community_w [64][128]
    float* sWe = smem + 8192  + wave * 2048;            // wave tile: w_e
    float* sA  = smem + 24576 + wave * 2048;            // wave tile: w_e * c_e

    // ---- cooperative load of community_w into LDS ([K][D] row-major as-is)
    {
        const float4* src = (const float4*)comm_w;
        float4*       dst = (float4*)sW;
        #pragma unroll
        for (int i = 0; i < (KCAT * DIM) / (4 * 256); ++i)
            dst[tid + i * 256] = src[tid + i * 256];
    }

    const int rowBase = blockIdx.x * ROWS_WG + wave * 16;

    // ---- stage the 16x128 A-tiles: one full embedding row per b128 wave load
    for (int r = 0; r < 16; ++r) {
        const int iw = w_idx[rowBase + r];
        const int ic = c_idx[rowBase + r];
        float4 we = *(const float4*)(node_emb + (size_t)iw * DIM + lane * 4);
        float4 ce = *(const float4*)(node_emb + (size_t)ic * DIM + lane * 4);
        float4 pr;
        pr.x = we.x * ce.x; pr.y = we.y * ce.y;
        pr.z = we.z * ce.z; pr.w = we.w * ce.w;
        *(float4*)(sWe + r * DIM + lane * 4) = we;
        *(float4*)(sA  + r * DIM + lane * 4) = pr;
    }

    __syncthreads();   // community_w visible to all waves

    // ---- dual GEMM via V_WMMA_F32_16X16X4_F32:
    //      q     = (w_e*c_e)[16x128] x Wt[128x64]
    //      prior =  w_e     [16x128] x Wt[128x64]      (Wt[d][n] = comm_w[n][d])
    v8f accQ[4] = {};
    v8f accP[4] = {};

    const int m    = lane & 15;
    const int koff = (lane < 16) ? 0 : 2;
    const float* aArow = sA  + m * DIM;
    const float* aWrow = sWe + m * DIM;

    #pragma unroll 4
    for (int kt = 0; kt < 32; ++kt) {
        const int k0 = kt * 4 + koff;
        v2f aA = *(const v2f*)(aArow + k0);      // ds_load_b64
        v2f aW = *(const v2f*)(aWrow + k0);      // ds_load_b64
        #pragma unroll
        for (int nt = 0; nt < 4; ++nt) {
            v2f bf = *(const v2f*)(sW + (nt * 16 + m) * DIM + k0);   // ds_load_b64
            accQ[nt] = __builtin_amdgcn_wmma_f32_16x16x4_f32(
                false, aA, false, bf, (short)0, accQ[nt], false, false);
            accP[nt] = __builtin_amdgcn_wmma_f32_16x16x4_f32(
                false, aW, false, bf, (short)0, accP[nt], false, false);
        }
    }

    // ---- spill accumulators to LDS as [row 0..15][64], reusing staging tiles
    // C/D layout: VGPR r, lane L -> row = r + (L<16?0:8), col = nt*16 + (L%16)
    const int mbase = (lane < 16) ? 0 : 8;
    #pragma unroll
    for (int nt = 0; nt < 4; ++nt) {
        #pragma unroll
        for (int r = 0; r < 8; ++r) {
            sA [(mbase + r) * KCAT + nt * 16 + m] = accQ[nt][r];
            sWe[(mbase + r) * KCAT + nt * 16 + m] = accP[nt][r];
        }
    }
    // wave-private LDS area + in-order LDS per wave: no barrier needed.

    // ---- per-row phase: lanes 0..15 each own one batch row of this wave
    float contrib = 0.0f;
    if (lane < 16) {
        const int    row = rowBase + lane;
        const float* qr  = sA  + lane * KCAT;
        const float* pr  = sWe + lane * KCAT;
        const float  invT = 1.0f / (float)temp_p[0];

        // softmax(q) -> out_q
        float mx = qr[0];
        for (int k = 1; k < KCAT; ++k) mx = fmaxf(mx, qr[k]);
        float se = 0.0f;
        for (int k = 0; k < KCAT; ++k) se += __expf(qr[k] - mx);
        const float ise = 1.0f / se;
        float* oq = out_q + (size_t)row * KCAT;
        for (int k = 0; k < KCAT; ++k) oq[k] = __expf(qr[k] - mx) * ise;

        // prior = softmax(lin(w_e)) -> out_prior
        float mp = pr[0];
        for (int k = 1; k < KCAT; ++k) mp = fmaxf(mp, pr[k]);
        float sp = 0.0f;
        for (int k = 0; k < KCAT; ++k) sp += __expf(pr[k] - mp);
        const float isp = 1.0f / sp;
        float* op = out_prior + (size_t)row * KCAT;
        for (int k = 0; k < KCAT; ++k) op[k] = __expf(pr[k] - mp) * isp;

        // gumbel-softmax: argmax (first max, like jnp.argmax) + y_soft[kmax]
        const float* gr = gumbel + (size_t)row * KCAT;
        float smax = -INFINITY; int kmax = 0;
        for (int k = 0; k < KCAT; ++k) {
            float sv = (qr[k] + gr[k]) * invT;
            if (sv > smax) { smax = sv; kmax = k; }
        }
        float ssum = 0.0f;
        for (int k = 0; k < KCAT; ++k) ssum += __expf((qr[k] + gr[k]) * invT - smax);
        const float ysm  = 1.0f / ssum;            // softmax value at argmax
        const float zmax = (1.0f - ysm) + ysm;     // ST estimator rounding, faithfully

        // z is exactly one-hot except at kmax => Σ z*lin(x) = zmax * dot(x, W[kmax])
        const float* Wk = sW + kmax * DIM;

        {   // positive term
            const float* cr = ctx_emb + (size_t)c_idx[row] * DIM;
            float d = 0.0f;
            for (int k4 = 0; k4 < DIM; k4 += 4) {
                float4 a = *(const float4*)(cr + k4);
                float4 b = *(const float4*)(Wk + k4);
                d += a.x * b.x + a.y * b.y + a.z * b.z + a.w * b.w;
            }
            contrib += log_sigmoid(zmax * d);
        }
        float nsum = 0.0f;
        for (int n = 0; n < NNEG; ++n) {   // negative terms
            const float* nr = ctx_emb + (size_t)neg_idx[row * NNEG + n] * DIM;
            float d = 0.0f;
            for (int k4 = 0; k4 < DIM; k4 += 4) {
                float4 a = *(const float4*)(nr + k4);
                float4 b = *(const float4*)(Wk + k4);
                d += a.x * b.x + a.y * b.y + a.z * b.z + a.w * b.w;
            }
            nsum += log_sigmoid(-(zmax * d));
        }
        contrib += nsum * (1.0f / NNEG);
    }

    // ---- deterministic in-workgroup reduction (fixed order, no float atomics)
    red[tid] = contrib;
    __syncthreads();
    if (tid == 0) {
        float s = 0.0f;
        for (int i = 0; i < 256; ++i) s += red[i];
        loss_partials[blockIdx.x] = s;
    }
}

__global__ __launch_bounds__(256)
void gcn_gumbel_reduce(const float* __restrict__ partials, int n, float invB,
                       float* __restrict__ out_loss)
{
    __shared__ float red[256];
    const int t = threadIdx.x;
    float s = 0.0f;
    for (int i = t; i < n; i += 256) s += partials[i];   // fixed per-thread order
    red[t] = s;
    __syncthreads();
    if (t == 0) {
        float tot = 0.0f;
        for (int i = 0; i < 256; ++i) tot += red[i];     // fixed order
        out_loss[0] = -tot * invB;
    }
}

extern "C" void kernel_launch(void* const* d_in, const int* in_sizes, int n_in,
                              void* d_out, int out_size, void* d_ws, size_t ws_size,
                              hipStream_t stream) {
    (void)n_in; (void)out_size; (void)ws_size;
    const int*   w    = (const int*)  d_in[0];
    const int*   c    = (const int*)  d_in[1];
    const int*   neg  = (const int*)  d_in[2];
    const int*   temp = (const int*)  d_in[3];
    const float* gn   = (const float*)d_in[4];
    const float* ne   = (const float*)d_in[5];
    const float* ce   = (const float*)d_in[6];
    const float* cw   = (const float*)d_in[7];

    const int B       = in_sizes[0];            // 131072
    const int nblocks = B / ROWS_WG;            // 1024

    float* out      = (float*)d_out;            // [0]=loss, then q_softmax, then prior
    float* partials = (float*)d_ws;             // nblocks floats of scratch

    gcn_gumbel_main<<<nblocks, 256, 0, stream>>>(
        w, c, neg, temp, gn, ne, ce, cw,
        partials, out + 1, out + 1 + (size_t)B * KCAT);

    gcn_gumbel_reduce<<<1, 256, 0, stream>>>(partials, nblocks, 1.0f / (float)B, out);
}